// MyModel_3633542333209
// MI455X (gfx1250) — compile-verified
//
#include <hip/hip_runtime.h>
#include <math.h>

// ---------------------------------------------------------------------------
// MI455X / gfx1250 implementation.  FP32 WMMA (v_wmma_f32_16x16x4_f32) for all
// GEMMs with 16x64 register tiling (one A fragment feeds 4 WMMAs); atomics-
// based graph aggregation kept resident in the 192MB L2.
// ---------------------------------------------------------------------------

typedef __attribute__((ext_vector_type(2))) float v2f;
typedef __attribute__((ext_vector_type(8))) float v8f;

#define BN_EPS 1e-5f

// ------------------------------ WMMA GEMM ----------------------------------
// C[M,N] = A[M,K] @ B[K,N] + bias[N]  (optional ReLU).  Row-major everything.
// One 16x64 output strip per wave (4 accumulators over 4 consecutive N-tiles);
// 8 waves per 256-thread block (wave32).  A fragment (16x4 f32): lanes 0-15
// hold M=0..15 with K={k0,k0+1} in two VGPRs, lanes 16-31 hold K={k0+2,k0+3}.
// B fragment (4x16) mirrors that with N on lanes.  C/D: VGPR r -> row r
// (lanes 0-15) / row r+8 (lanes 16-31).
__global__ void k_gemm_bias_wmma(const float* __restrict__ A,
                                 const float* __restrict__ B,
                                 const float* __restrict__ bias,
                                 float* __restrict__ C,
                                 int M, int N, int K, int relu) {
  const int lane = threadIdx.x & 31;
  const int wave = threadIdx.x >> 5;
  const int ngrp   = N >> 6;          // groups of 4 N-tiles (64 cols)
  const int mtiles = M >> 4;
  int tile = blockIdx.x * (blockDim.x >> 5) + wave;
  if (tile >= mtiles * ngrp) return;  // wave-uniform: EXEC stays all-1s
  const int mt = tile / ngrp;
  const int ng = tile % ngrp;

  const int lo  = lane & 15;          // M index (A) / N index (B,C)
  const int kof = (lane >> 4) * 2;    // K sub-offset for this half-wave

  const float* Arow = A + (size_t)(mt * 16 + lo) * K;
  const float* Bcol = B + ng * 64 + lo;

  v8f acc0 = {}, acc1 = {}, acc2 = {}, acc3 = {};
  for (int k0 = 0; k0 < K; k0 += 4) {
    v2f a;
    a.x = Arow[k0 + kof];
    a.y = Arow[k0 + kof + 1];
    const float* Br0 = Bcol + (size_t)(k0 + kof) * N;
    const float* Br1 = Bcol + (size_t)(k0 + kof + 1) * N;
    v2f b0, b1, b2, b3;
    b0.x = Br0[0];  b0.y = Br1[0];
    b1.x = Br0[16]; b1.y = Br1[16];
    b2.x = Br0[32]; b2.y = Br1[32];
    b3.x = Br0[48]; b3.y = Br1[48];
    acc0 = __builtin_amdgcn_wmma_f32_16x16x4_f32(false, a, false, b0, (short)0, acc0, false, false);
    acc1 = __builtin_amdgcn_wmma_f32_16x16x4_f32(false, a, false, b1, (short)0, acc1, false, false);
    acc2 = __builtin_amdgcn_wmma_f32_16x16x4_f32(false, a, false, b2, (short)0, acc2, false, false);
    acc3 = __builtin_amdgcn_wmma_f32_16x16x4_f32(false, a, false, b3, (short)0, acc3, false, false);
  }

  const int mbase = mt * 16 + (lane >> 4) * 8;
  v8f accs[4] = {acc0, acc1, acc2, acc3};
#pragma unroll
  for (int nt = 0; nt < 4; ++nt) {
    const int nc   = ng * 64 + nt * 16 + lo;
    const float bv = bias ? bias[nc] : 0.0f;
#pragma unroll
    for (int r = 0; r < 8; ++r) {
      float v = accs[nt][r] + bv;
      if (relu) v = fmaxf(v, 0.0f);
      C[(size_t)(mbase + r) * N + nc] = v;
    }
  }
}

static inline void launch_gemm(const float* A, const float* B, const float* bias,
                               float* C, int M, int N, int K, int relu,
                               hipStream_t s) {
  int tiles  = (M >> 4) * (N >> 6);
  int blocks = (tiles + 7) / 8;
  k_gemm_bias_wmma<<<blocks, 256, 0, s>>>(A, B, bias, C, M, N, K, relu);
}

// ------------------------------ utilities ----------------------------------
__global__ void k_fill(float* __restrict__ p, float v, int n) {
  int i = blockIdx.x * blockDim.x + threadIdx.x;
  if (i < n) p[i] = v;
}

__global__ void k_edge_deg(const int* __restrict__ dst, float* __restrict__ deg, int E) {
  int e = blockIdx.x * blockDim.x + threadIdx.x;
  if (e < E) atomicAdd(&deg[dst[e]], 1.0f);
}

__global__ void k_rsqrt(const float* __restrict__ deg, float* __restrict__ dinv, int n) {
  int i = blockIdx.x * blockDim.x + threadIdx.x;
  if (i < n) dinv[i] = rsqrtf(deg[i]);
}

// out[i,f] = h[i,f] / deg[i]   (self-loop term, norm 1/deg)
__global__ void k_selfinit(const float* __restrict__ h, const float* __restrict__ deg,
                           float* __restrict__ out, int N) {
  int i = blockIdx.x, f = threadIdx.x;
  out[(size_t)i * N + f] = h[(size_t)i * N + f] / deg[i];
}

// out[dst,f] += dinv[src]*dinv[dst]*h[src,f]
__global__ void k_edge_agg(const float* __restrict__ h, const float* __restrict__ dinv,
                           const int* __restrict__ src, const int* __restrict__ dst,
                           float* __restrict__ out, int N) {
  int e = blockIdx.x, f = threadIdx.x;
  int s = src[e], d = dst[e];
  float w = dinv[s] * dinv[d];
  atomicAdd(&out[(size_t)d * N + f], w * h[(size_t)s * N + f]);
}

// ------------------------------ batch norm ---------------------------------
__global__ void k_bn_stats(const float* __restrict__ X, float* __restrict__ mean,
                           float* __restrict__ var, int M, int N) {
  int col = blockIdx.x;
  float s = 0.f, ss = 0.f;
  for (int r = threadIdx.x; r < M; r += blockDim.x) {
    float v = X[(size_t)r * N + col];
    s += v; ss += v * v;
  }
  __shared__ float sh[256], sh2[256];
  sh[threadIdx.x] = s; sh2[threadIdx.x] = ss;
  __syncthreads();
  for (int st = blockDim.x >> 1; st > 0; st >>= 1) {
    if ((int)threadIdx.x < st) {
      sh[threadIdx.x]  += sh[threadIdx.x + st];
      sh2[threadIdx.x] += sh2[threadIdx.x + st];
    }
    __syncthreads();
  }
  if (threadIdx.x == 0) {
    float m = sh[0] / (float)M;
    mean[col] = m;
    var[col]  = sh2[0] / (float)M - m * m;   // biased variance
  }
}

__global__ void k_bn_apply_relu(float* __restrict__ X, const float* __restrict__ mean,
                                const float* __restrict__ var, const float* __restrict__ g,
                                const float* __restrict__ be, int N) {
  int i = blockIdx.x, f = threadIdx.x;
  size_t idx = (size_t)i * N + f;
  float v = X[idx];
  v = g[f] * (v - mean[f]) * rsqrtf(var[f] + BN_EPS) + be[f];
  X[idx] = fmaxf(v, 0.0f);
}

// ------------------------------ mol front end ------------------------------
// h[i,:] = sum_f emb[f, mol_x[i,f], :]      emb: [NF=9, V=128, H=256]
__global__ void k_atom_enc(const int* __restrict__ mol_x, const float* __restrict__ emb,
                           float* __restrict__ h) {
  int i = blockIdx.x, f = threadIdx.x;            // f < 256
  float s = 0.f;
#pragma unroll
  for (int j = 0; j < 9; ++j) {
    int xv = mol_x[i * 9 + j];
    s += emb[((size_t)j * 128 + xv) * 256 + f];
  }
  h[(size_t)i * 256 + f] = s;
}

__global__ void k_pool(const float* __restrict__ mfeat, const int* __restrict__ batch,
                       float* __restrict__ mpool) {
  int i = blockIdx.x, f = threadIdx.x;            // f < 128
  atomicAdd(&mpool[(size_t)batch[i] * 128 + f], mfeat[(size_t)i * 128 + f]);
}

// ------------------------------ head ---------------------------------------
__global__ void k_concat(const float* __restrict__ dfeat, const int* __restrict__ didx,
                         const float* __restrict__ mpool,
                         const float* __restrict__ sfeat, const int* __restrict__ sidx,
                         float* __restrict__ fea) {
  int b = blockIdx.x, f = threadIdx.x;            // f < 128
  size_t o = (size_t)b * 384;
  fea[o + f]       = dfeat[(size_t)didx[b] * 128 + f];
  fea[o + 128 + f] = mpool[(size_t)b * 128 + f];
  fea[o + 256 + f] = sfeat[(size_t)sidx[b] * 128 + f];
}

// conv1d(k=3, pad=1) -> maxpool(2,2).  threads = Cout*Lout per block.
__global__ void k_conv_pool(const float* __restrict__ x, const float* __restrict__ W,
                            const float* __restrict__ bias, float* __restrict__ y,
                            int Cin, int Cout, int Lin) {
  int b = blockIdx.x;
  int Lout = Lin >> 1;
  int o = threadIdx.x / Lout;
  int t = threadIdx.x % Lout;
  float m = -3.4e38f;
#pragma unroll
  for (int p = 0; p < 2; ++p) {
    int tt = 2 * t + p;
    float acc = bias[o];
    for (int i = 0; i < Cin; ++i)
      for (int dk = 0; dk < 3; ++dk) {
        int xt = tt + dk - 1;
        if (xt >= 0 && xt < Lin)
          acc += W[(o * Cin + i) * 3 + dk] * x[((size_t)b * Cin + i) * Lin + xt];
      }
    m = fmaxf(m, acc);
  }
  y[((size_t)b * Cout + o) * Lout + t] = m;
}

__global__ void k_bn_ch_stats(const float* __restrict__ y, float* __restrict__ mean,
                              float* __restrict__ var, int C, int L, int Bn) {
  int c = blockIdx.x;
  float s = 0.f, ss = 0.f;
  for (int idx = threadIdx.x; idx < Bn * L; idx += blockDim.x) {
    int b = idx / L, t = idx % L;
    float v = y[((size_t)b * C + c) * L + t];
    s += v; ss += v * v;
  }
  __shared__ float sh[256], sh2[256];
  sh[threadIdx.x] = s; sh2[threadIdx.x] = ss;
  __syncthreads();
  for (int st = blockDim.x >> 1; st > 0; st >>= 1) {
    if ((int)threadIdx.x < st) {
      sh[threadIdx.x]  += sh[threadIdx.x + st];
      sh2[threadIdx.x] += sh2[threadIdx.x + st];
    }
    __syncthreads();
  }
  if (threadIdx.x == 0) {
    float m = sh[0] / (float)(Bn * L);
    mean[c] = m;
    var[c]  = sh2[0] / (float)(Bn * L) - m * m;
  }
}

__global__ void k_bn_ch_apply_relu(float* __restrict__ y, const float* __restrict__ mean,
                                   const float* __restrict__ var, const float* __restrict__ g,
                                   const float* __restrict__ be, int C, int L) {
  int b = blockIdx.x;
  int c = threadIdx.x / L;
  int t = threadIdx.x % L;
  size_t idx = ((size_t)b * C + c) * L + t;
  float v = y[idx];
  v = g[c] * (v - mean[c]) * rsqrtf(var[c] + BN_EPS) + be[c];
  y[idx] = fmaxf(v, 0.0f);
}

__global__ void k_lin2(const float* __restrict__ x, const float* __restrict__ W,
                       const float* __restrict__ bias, float* __restrict__ out, int Bn, int K) {
  int b = blockIdx.x * blockDim.x + threadIdx.x;
  if (b >= Bn) return;
  float acc = bias[0];
  for (int k = 0; k < K; ++k) acc += x[(size_t)b * K + k] * W[k];
  out[b] = acc;
}

// ------------------------------ host orchestration -------------------------
static void run_gcn(const float* x0, int n, int d_in,
                    const int* src, const int* dst, int E,
                    const float* W0, const float* W1, const float* W2,
                    const float* b0, const float* b1, const float* b2,
                    const float* g0, const float* g1,
                    const float* be0, const float* be1,
                    float* bufH, float* bufO, float* bufX2,
                    float* deg, float* dinv, float* mean, float* var,
                    float* feat_out, hipStream_t s) {
  const int H = 256, OUT = 128;
  // degrees (include self-loop)
  k_fill<<<(n + 255) / 256, 256, 0, s>>>(deg, 1.0f, n);
  k_edge_deg<<<(E + 255) / 256, 256, 0, s>>>(dst, deg, E);
  k_rsqrt<<<(n + 255) / 256, 256, 0, s>>>(deg, dinv, n);

  // layer 0: d_in -> H, BN+ReLU
  launch_gemm(x0, W0, b0, bufH, n, H, d_in, 0, s);
  k_selfinit<<<n, H, 0, s>>>(bufH, deg, bufO, H);
  k_edge_agg<<<E, H, 0, s>>>(bufH, dinv, src, dst, bufO, H);
  k_bn_stats<<<H, 256, 0, s>>>(bufO, mean, var, n, H);
  k_bn_apply_relu<<<n, H, 0, s>>>(bufO, mean, var, g0, be0, H);

  // layer 1: H -> H, BN+ReLU
  launch_gemm(bufO, W1, b1, bufH, n, H, H, 0, s);
  k_selfinit<<<n, H, 0, s>>>(bufH, deg, bufX2, H);
  k_edge_agg<<<E, H, 0, s>>>(bufH, dinv, src, dst, bufX2, H);
  k_bn_stats<<<H, 256, 0, s>>>(bufX2, mean, var, n, H);
  k_bn_apply_relu<<<n, H, 0, s>>>(bufX2, mean, var, g1, be1, H);

  // layer 2: H -> OUT (no BN/ReLU)
  launch_gemm(bufX2, W2, b2, bufH, n, OUT, H, 0, s);
  k_selfinit<<<n, OUT, 0, s>>>(bufH, deg, feat_out, OUT);
  k_edge_agg<<<E, OUT, 0, s>>>(bufH, dinv, src, dst, feat_out, OUT);
}

extern "C" void kernel_launch(void* const* d_in, const int* in_sizes, int n_in,
                              void* d_out, int out_size, void* d_ws, size_t ws_size,
                              hipStream_t stream) {
  (void)in_sizes; (void)n_in; (void)out_size; (void)ws_size;
  const int N_DRUG = 20000, N_DIS = 20000, E_G = 320000;
  const int Bsz = 4096, N_MOL = 65536, E_MOL = 262144;

  // ---- inputs (setup_inputs dict order) ----
  const float* drug_x   = (const float*)d_in[0];
  const int*   drug_ei  = (const int*)d_in[1];
  const int*   drug_idx = (const int*)d_in[2];
  const float* dis_x    = (const float*)d_in[3];
  const int*   dis_ei   = (const int*)d_in[4];
  const int*   dis_idx  = (const int*)d_in[5];
  const int*   mol_x    = (const int*)d_in[6];
  const int*   mol_ei   = (const int*)d_in[7];
  const int*   mol_bat  = (const int*)d_in[8];
  const float* dr_W[3]  = {(const float*)d_in[9],  (const float*)d_in[10], (const float*)d_in[11]};
  const float* dr_b[3]  = {(const float*)d_in[12], (const float*)d_in[13], (const float*)d_in[14]};
  const float* dr_g[2]  = {(const float*)d_in[15], (const float*)d_in[16]};
  const float* dr_be[2] = {(const float*)d_in[17], (const float*)d_in[18]};
  const float* di_W[3]  = {(const float*)d_in[19], (const float*)d_in[20], (const float*)d_in[21]};
  const float* di_b[3]  = {(const float*)d_in[22], (const float*)d_in[23], (const float*)d_in[24]};
  const float* di_g[2]  = {(const float*)d_in[25], (const float*)d_in[26]};
  const float* di_be[2] = {(const float*)d_in[27], (const float*)d_in[28]};
  const float* mol_emb  = (const float*)d_in[29];
  const float* mo_W[3]  = {(const float*)d_in[30], (const float*)d_in[31], (const float*)d_in[32]};
  const float* mo_b[3]  = {(const float*)d_in[33], (const float*)d_in[34], (const float*)d_in[35]};
  const float* mo_g[2]  = {(const float*)d_in[36], (const float*)d_in[37]};
  const float* mo_be[2] = {(const float*)d_in[38], (const float*)d_in[39]};
  const float* c1_W = (const float*)d_in[40]; const float* c1_b = (const float*)d_in[41];
  const float* c1_g = (const float*)d_in[42]; const float* c1_be = (const float*)d_in[43];
  const float* c2_W = (const float*)d_in[44]; const float* c2_b = (const float*)d_in[45];
  const float* c2_g = (const float*)d_in[46]; const float* c2_be = (const float*)d_in[47];
  const float* l1_W = (const float*)d_in[48]; const float* l1_b = (const float*)d_in[49];
  const float* l2_W = (const float*)d_in[50]; const float* l2_b = (const float*)d_in[51];
  float* out = (float*)d_out;

  // ---- workspace layout ----
  size_t off = 0;
  auto alloc = [&](size_t nelem) {
    float* p = (float*)d_ws + off;
    off += (nelem + 63) & ~(size_t)63;
    return p;
  };
  float* bufH  = alloc((size_t)N_MOL * 256);
  float* bufO  = alloc((size_t)N_MOL * 256);
  float* bufX2 = alloc((size_t)N_MOL * 256);
  float* deg   = alloc(N_MOL);
  float* dinv  = alloc(N_MOL);
  float* mean  = alloc(256);
  float* var   = alloc(256);
  float* dfeat = alloc((size_t)N_DRUG * 128);
  float* sfeat = alloc((size_t)N_DIS * 128);
  float* mfeat = alloc((size_t)N_MOL * 128);
  float* mpool = alloc((size_t)Bsz * 128);
  float* fea   = alloc((size_t)Bsz * 3 * 128);
  float* y1    = alloc((size_t)Bsz * 6 * 64);
  float* y2    = alloc((size_t)Bsz * 6 * 32);
  float* hid   = alloc((size_t)Bsz * 128);
  float* cmean = alloc(64);
  float* cvar  = alloc(64);

  // ---- drug GCN ----
  run_gcn(drug_x, N_DRUG, 128, drug_ei, drug_ei + E_G, E_G,
          dr_W[0], dr_W[1], dr_W[2], dr_b[0], dr_b[1], dr_b[2],
          dr_g[0], dr_g[1], dr_be[0], dr_be[1],
          bufH, bufO, bufX2, deg, dinv, mean, var, dfeat, stream);

  // ---- disease GCN ----
  run_gcn(dis_x, N_DIS, 128, dis_ei, dis_ei + E_G, E_G,
          di_W[0], di_W[1], di_W[2], di_b[0], di_b[1], di_b[2],
          di_g[0], di_g[1], di_be[0], di_be[1],
          bufH, bufO, bufX2, deg, dinv, mean, var, sfeat, stream);

  // ---- molecule GCN (atom encoder feeds layer 0) ----
  float* molh = bufO;  // temp reuse: layer-0 output; atom encoder writes bufX2
  k_atom_enc<<<N_MOL, 256, 0, stream>>>(mol_x, mol_emb, bufX2);
  run_gcn(bufX2, N_MOL, 256, mol_ei, mol_ei + E_MOL, E_MOL,
          mo_W[0], mo_W[1], mo_W[2], mo_b[0], mo_b[1], mo_b[2],
          mo_g[0], mo_g[1], mo_be[0], mo_be[1],
          bufH, molh, bufX2, deg, dinv, mean, var, mfeat, stream);
  // NOTE: run_gcn's layer-1 output overwrites bufX2 only after layer-0 GEMM
  // has consumed the atom-encoder features, so the reuse is safe.

  // ---- global_add_pool ----
  k_fill<<<(Bsz * 128 + 255) / 256, 256, 0, stream>>>(mpool, 0.0f, Bsz * 128);
  k_pool<<<N_MOL, 128, 0, stream>>>(mfeat, mol_bat, mpool);

  // ---- concat -> [B,3,128] ----
  k_concat<<<Bsz, 128, 0, stream>>>(dfeat, drug_idx, mpool, sfeat, dis_idx, fea);

  // ---- conv block 1: [B,3,128] -> [B,6,64], BN+ReLU ----
  k_conv_pool<<<Bsz, 6 * 64, 0, stream>>>(fea, c1_W, c1_b, y1, 3, 6, 128);
  k_bn_ch_stats<<<6, 256, 0, stream>>>(y1, cmean, cvar, 6, 64, Bsz);
  k_bn_ch_apply_relu<<<Bsz, 6 * 64, 0, stream>>>(y1, cmean, cvar, c1_g, c1_be, 6, 64);

  // ---- conv block 2: [B,6,64] -> [B,6,32], BN+ReLU ----
  k_conv_pool<<<Bsz, 6 * 32, 0, stream>>>(y1, c2_W, c2_b, y2, 6, 6, 64);
  k_bn_ch_stats<<<6, 256, 0, stream>>>(y2, cmean, cvar, 6, 32, Bsz);
  k_bn_ch_apply_relu<<<Bsz, 6 * 32, 0, stream>>>(y2, cmean, cvar, c2_g, c2_be, 6, 32);

  // ---- linear 1 (WMMA GEMM, fused ReLU): [B,192] @ [192,128] ----
  launch_gemm(y2, l1_W, l1_b, hid, Bsz, 128, 192, 1, stream);

  // ---- linear 2: [B,128] @ [128,1] ----
  k_lin2<<<(Bsz + 255) / 256, 256, 0, stream>>>(hid, l2_W, l2_b, out, Bsz, 128);
}